// NeighborhoodAttention2D_21388937134559
// MI455X (gfx1250) — compile-verified
//
#include <hip/hip_runtime.h>

// ---------------------------------------------------------------------------
// NeighborhoodAttention2D for MI455X (gfx1250), all-WMMA fp32 pipeline.
//   Stage 1: QKV GEMM  -> q (pre-scaled), k/v into zero-padded spatial arrays
//   Stage 2: FNA-style window attention, 1 wave per (b, head, 2x8 query tile)
//   Stage 3: output projection GEMM
// Matrix op: V_WMMA_F32_16X16X4_F32  (D = A(16x4) * B(4x16) + C, f32 all)
// ---------------------------------------------------------------------------

typedef __attribute__((ext_vector_type(2))) float v2f;
typedef __attribute__((ext_vector_type(4))) float v4f;
typedef __attribute__((ext_vector_type(8))) float v8f;

#define NHEADS 4
#define HDIM   32
#define IMH    56
#define IMW    56
#define HWPIX  3136      // 56*56
#define BATCH  4
#define PADW   62        // 56 + 2*3
#define QSCALE 0.17677669529663687f   // 32^-0.5

static __device__ __forceinline__ v8f wmma4(v2f a, v2f b, v8f c) {
  // 8-arg pattern: (neg_a, A, neg_b, B, c_mod, C, reuse_a, reuse_b)
  return __builtin_amdgcn_wmma_f32_16x16x4_f32(false, a, false, b, (short)0, c,
                                               false, false);
}

// ---------------------------------------------------------------------------
// Kernel 1: QKV projection. One block = one 16-pixel M-strip, 8 waves,
// each wave computes 3 of the 24 output 16-channel tiles.
// ---------------------------------------------------------------------------
__global__ __launch_bounds__(256) void qkv_kernel(
    const float* __restrict__ x, const float* __restrict__ qkv_w,
    const float* __restrict__ qkv_b, float* __restrict__ q_arr,
    float* __restrict__ kpad, float* __restrict__ vpad) {
  __shared__ float Xs[16 * 132];  // 16 pixels x 128 ch, row pad 132 (bank-spread)
  const int strip = blockIdx.x;
  const int t = threadIdx.x;

  for (int idx = t; idx < 16 * 32; idx += 256) {  // 512 float4 total
    int m = idx >> 5, q4 = idx & 31;
    *(v4f*)&Xs[m * 132 + q4 * 4] =
        *(const v4f*)&x[(size_t)(strip * 16 + m) * 128 + q4 * 4];
  }
  __syncthreads();

  const int lane = t & 31;
  const int half = lane >> 4;    // K-subrange select per ISA A/B layouts
  const int n16  = lane & 15;    // M for A-frags, N for B-frags
  const int wave = t >> 5;

  for (int tt = 0; tt < 3; ++tt) {
    const int otile = wave * 3 + tt;
    const int obase = otile * 16;
    const int o = obase + n16;
    __builtin_prefetch(&qkv_w[(size_t)o * 128], 0, 1);
    v8f acc = {};
#pragma unroll
    for (int kc = 0; kc < 32; ++kc) {
      const int ch = kc * 4 + 2 * half;
      v2f a = *(const v2f*)&Xs[n16 * 132 + ch];               // A: X[m][k..k+1]
      v2f b = *(const v2f*)&qkv_w[(size_t)o * 128 + ch];      // B: W[o][k..k+1]
      acc = wmma4(a, b, acc);
    }
    const float bias = qkv_b[o];
    const int which = otile >> 3;          // 0=q 1=k 2=v
    const int head  = (otile & 7) >> 1;
    const int d     = (otile & 1) * 16 + n16;
#pragma unroll
    for (int vr = 0; vr < 8; ++vr) {
      const int m = vr + 8 * half;                 // C/D row
      const unsigned pixel = strip * 16 + m;
      const unsigned bimg = pixel / HWPIX;
      const unsigned rem  = pixel % HWPIX;
      const unsigned i = rem / IMW, j = rem % IMW;
      const float val = acc[vr] + bias;
      if (which == 0) {
        q_arr[(((size_t)bimg * NHEADS + head) * HWPIX + rem) * HDIM + d] =
            val * QSCALE;
      } else {
        float* dst = (which == 1) ? kpad : vpad;
        dst[((((size_t)bimg * NHEADS + head) * PADW + (i + 3)) * PADW +
             (j + 3)) * HDIM + d] = val;
      }
    }
  }
}

// ---------------------------------------------------------------------------
// Kernel 2: window attention. 1 wave per (b, head, 2x8 query tile).
// Key union block = 8 rows x 14 cols = 112 positions.
//   S^T = Kblk(112x32) * Q^T(32x16)  -> 7 WMMA tiles x 8 K-chunks
//   softmax per query (lane pair, one shfl_xor 16)
//   out^T = V^T(32x112) * P^T(112x16) -> 2 WMMA tiles x 28 K-chunks
// ---------------------------------------------------------------------------
__global__ __launch_bounds__(32) void attn_kernel(
    const float* __restrict__ q_arr, const float* __restrict__ kpad,
    const float* __restrict__ vpad, const float* __restrict__ rpb,
    float* __restrict__ attnout) {
  __shared__ float Ks[112 * 36];   // key block, row pad 36 (16B aligned, bank-spread)
  __shared__ float Vs[112 * 36];
  __shared__ float BiasT[49];
  float* PT = Ks;                  // reuse K buffer for P^T (112 x 17)

  int unit = blockIdx.x;
  const int tj = unit % 7;  unit /= 7;
  const int ti = unit % 28; unit /= 28;
  const int h  = unit % NHEADS; unit /= NHEADS;
  const int b  = unit;
  const int i0 = ti * 2, j0 = tj * 8;

  const int lane = threadIdx.x;
  const int half = lane >> 4;
  const int n16  = lane & 15;

  for (int idx = lane; idx < 49; idx += 32) {
    const int di = idx / 7, dj = idx % 7;
    BiasT[idx] = rpb[((size_t)h * 13 + (di + 3)) * 13 + (dj + 3)];
  }

  const size_t kvrow0 = (((size_t)(b * NHEADS + h)) * PADW + i0) * PADW;
  for (int idx = lane; idx < 112 * 8; idx += 32) {  // 112 pos x 8 float4
    const int pos = idx >> 3, q4 = idx & 7;
    const int kr = pos / 14, kc = pos % 14;
    const size_t g = (kvrow0 + kr * PADW + (j0 + kc)) * HDIM + q4 * 4;
    *(v4f*)&Ks[pos * 36 + q4 * 4] = *(const v4f*)&kpad[g];
    *(v4f*)&Vs[pos * 36 + q4 * 4] = *(const v4f*)&vpad[g];
  }

  // Q^T B-fragments (query n16, already scaled by QSCALE in kernel 1)
  const int r = n16 >> 3, cq = n16 & 7;
  const int qi = i0 + r, qj = j0 + cq;
  const size_t qbase =
      (((size_t)(b * NHEADS + h)) * HWPIX + qi * IMW + qj) * HDIM;
  v2f qb[8];
#pragma unroll
  for (int kc = 0; kc < 8; ++kc)
    qb[kc] = *(const v2f*)&q_arr[qbase + kc * 4 + 2 * half];
  __syncthreads();

  // --- matmul 1: S^T tiles ---
  v8f S[7];
#pragma unroll
  for (int tt = 0; tt < 7; ++tt) {
    v8f acc = {};
    const int pos = tt * 16 + n16;  // A row = key position
#pragma unroll
    for (int kc = 0; kc < 8; ++kc) {
      v2f a = *(const v2f*)&Ks[pos * 36 + kc * 4 + 2 * half];
      acc = wmma4(a, qb[kc], acc);
    }
#pragma unroll
    for (int vr = 0; vr < 8; ++vr) {  // bias + window mask
      const int key = tt * 16 + vr + 8 * half;
      const int kr = key / 14, kc2 = key % 14;
      const int di = kr - r, dj = kc2 - cq;
      if (di >= 0 && di < 7 && dj >= 0 && dj < 7)
        acc[vr] += BiasT[di * 7 + dj];
      else
        acc[vr] = -1e30f;
    }
    S[tt] = acc;
  }

  // --- softmax over 112 keys (56 per lane + partner lane via xor16) ---
  float mx = -1e30f;
#pragma unroll
  for (int tt = 0; tt < 7; ++tt)
#pragma unroll
    for (int vr = 0; vr < 8; ++vr) mx = fmaxf(mx, S[tt][vr]);
  mx = fmaxf(mx, __shfl_xor(mx, 16, 32));
  float sum = 0.f;
#pragma unroll
  for (int tt = 0; tt < 7; ++tt)
#pragma unroll
    for (int vr = 0; vr < 8; ++vr) {
      const float e = __expf(S[tt][vr] - mx);
      S[tt][vr] = e;
      sum += e;
    }
  sum += __shfl_xor(sum, 16, 32);
  const float inv = 1.f / sum;

  __syncthreads();  // K-block reads done; safe to overwrite with P^T
#pragma unroll
  for (int tt = 0; tt < 7; ++tt)
#pragma unroll
    for (int vr = 0; vr < 8; ++vr)
      PT[(tt * 16 + vr + 8 * half) * 17 + n16] = S[tt][vr];
  __syncthreads();

  // --- matmul 2: out^T = V^T * P^T ---
  const size_t outbase =
      ((size_t)(b * HWPIX) + qi * IMW + qj) * 128 + h * HDIM;
#pragma unroll
  for (int mt = 0; mt < 2; ++mt) {
    v8f acc = {};
    const int chA = mt * 16 + n16;  // A row = output channel
#pragma unroll
    for (int kc = 0; kc < 28; ++kc) {
      const int k0 = kc * 4 + 2 * half;
      v2f a;
      a.x = Vs[k0 * 36 + chA];
      a.y = Vs[(k0 + 1) * 36 + chA];
      v2f bfr;
      bfr.x = PT[k0 * 17 + n16];
      bfr.y = PT[(k0 + 1) * 17 + n16];
      acc = wmma4(a, bfr, acc);
    }
#pragma unroll
    for (int vr = 0; vr < 8; ++vr)
      attnout[outbase + mt * 16 + vr + 8 * half] = acc[vr] * inv;
  }
}

// ---------------------------------------------------------------------------
// Kernel 3: output projection. One block = one 16-pixel strip, 8 waves,
// each wave one 16-channel output tile.
// ---------------------------------------------------------------------------
__global__ __launch_bounds__(256) void proj_kernel(
    const float* __restrict__ attnout, const float* __restrict__ proj_w,
    const float* __restrict__ proj_b, float* __restrict__ out) {
  __shared__ float Xs[16 * 132];
  const int strip = blockIdx.x;
  const int t = threadIdx.x;
  for (int idx = t; idx < 16 * 32; idx += 256) {
    int m = idx >> 5, q4 = idx & 31;
    *(v4f*)&Xs[m * 132 + q4 * 4] =
        *(const v4f*)&attnout[(size_t)(strip * 16 + m) * 128 + q4 * 4];
  }
  __syncthreads();
  const int lane = t & 31;
  const int half = lane >> 4;
  const int n16  = lane & 15;
  const int o = (t >> 5) * 16 + n16;
  __builtin_prefetch(&proj_w[(size_t)o * 128], 0, 1);
  v8f acc = {};
#pragma unroll
  for (int kc = 0; kc < 32; ++kc) {
    const int ch = kc * 4 + 2 * half;
    v2f a = *(const v2f*)&Xs[n16 * 132 + ch];
    v2f b = *(const v2f*)&proj_w[(size_t)o * 128 + ch];
    acc = wmma4(a, b, acc);
  }
  const float bias = proj_b[o];
#pragma unroll
  for (int vr = 0; vr < 8; ++vr) {
    const int m = vr + 8 * half;
    out[(size_t)(strip * 16 + m) * 128 + o] = acc[vr] + bias;
  }
}

// ---------------------------------------------------------------------------
extern "C" void kernel_launch(void* const* d_in, const int* in_sizes, int n_in,
                              void* d_out, int out_size, void* d_ws,
                              size_t ws_size, hipStream_t stream) {
  (void)in_sizes; (void)n_in; (void)out_size; (void)ws_size;
  const float* x      = (const float*)d_in[0];
  const float* qkv_w  = (const float*)d_in[1];
  const float* qkv_b  = (const float*)d_in[2];
  const float* rpb    = (const float*)d_in[3];
  const float* proj_w = (const float*)d_in[4];
  const float* proj_b = (const float*)d_in[5];

  float* ws = (float*)d_ws;
  const size_t q_len   = (size_t)BATCH * NHEADS * HWPIX * HDIM;      // 1.6M f
  const size_t pad_len = (size_t)BATCH * NHEADS * PADW * PADW * HDIM; // ~2.0M f
  float* q_arr   = ws;
  float* kpad    = ws + q_len;
  float* vpad    = kpad + pad_len;
  float* attnout = vpad + pad_len;

  // zero the padded K/V arrays (borders must read as 0, matching jnp.pad)
  hipMemsetAsync(kpad, 0, 2 * pad_len * sizeof(float), stream);

  qkv_kernel<<<784, 256, 0, stream>>>(x, qkv_w, qkv_b, q_arr, kpad, vpad);
  attn_kernel<<<BATCH * NHEADS * 28 * 7, 32, 0, stream>>>(q_arr, kpad, vpad,
                                                          rpb, attnout);
  proj_kernel<<<784, 256, 0, stream>>>(attnout, proj_w, proj_b, (float*)d_out);
}